// LocallyConnectedLayer_1675037245659
// MI455X (gfx1250) — compile-verified
//
#include <hip/hip_runtime.h>
#include <stdint.h>

typedef __attribute__((ext_vector_type(2))) float v2f;
typedef __attribute__((ext_vector_type(8))) float v8f;

#define BATCH   16
#define C_IN    32
#define C_OUT   64
#define KER     3
#define STRIDEH 2
#define H_OUT   56
#define W_OUT   56
#define H_IN    113
#define KTOT    (C_IN * KER * KER)   // 288

// One block per output position (u,v). 128 threads = 4 wave32s.
// Wave w computes D[b=0..15][o = w*16 .. w*16+15] via 72 V_WMMA_F32_16X16X4_F32.
__global__ __launch_bounds__(128)
void lcl_wmma_f32_kernel(const float* __restrict__ x,
                         const float* __restrict__ w,
                         const float* __restrict__ bias,
                         float* __restrict__ out)
{
    // A staged as [k][b]: row stride 16 dwords -> conflict-free WMMA-fragment reads
    __shared__ float Ald[KTOT][BATCH];   // 18432 bytes

    const int pos  = blockIdx.x;          // u*56 + v
    const int u    = pos / W_OUT;
    const int v    = pos % W_OUT;
    const int tid  = threadIdx.x;
    const int lane = tid & 31;
    const int wave = tid >> 5;

    // ---- async gather of the A tile into LDS (no VGPR round-trip, ASYNCcnt) ----
    // A[k][b] = x[b, c, 2u+q, 2v+r],  k = c*9 + q*3 + r
    const int row0 = u * STRIDEH;
    const int col0 = v * STRIDEH;
    for (int i = tid; i < KTOT * BATCH; i += 128) {   // 4608/128 = 36 iters, uniform
        const int b  = i & 15;
        const int k  = i >> 4;
        const int c  = k / (KER * KER);
        const int qr = k % (KER * KER);
        const int q  = qr / KER;
        const int r  = qr % KER;
        const float* gsrc = &x[((size_t)(b * C_IN + c) * H_IN + (row0 + q)) * H_IN + (col0 + r)];
        const uint32_t ldst = (uint32_t)(uintptr_t)&Ald[k][b];
        asm volatile("global_load_async_to_lds_b32 %0, %1, off"
                     :: "v"(ldst), "v"(gsrc) : "memory");
    }
    asm volatile("s_wait_asynccnt 0x0" ::: "memory");  // drain this wave's async copies
    __syncthreads();                                   // publish LDS to all 4 waves

    // ---- WMMA main loop: stream B (single-use weights) non-temporally ----
    const float* wbase = w + (size_t)pos * KTOT * C_OUT + wave * 16; // this wave's N-tile
    const int n     = lane & 15;          // M for A-frag, N for B/C/D-frags
    const int khalf = (lane >> 4) << 1;   // lanes 16..31 hold K = k0+2, k0+3

    v8f acc = {};
    #pragma unroll 8
    for (int k0 = 0; k0 < KTOT; k0 += 4) {
        v2f a, b;
        a.x = Ald[k0 + khalf + 0][n];
        a.y = Ald[k0 + khalf + 1][n];
        b.x = __builtin_nontemporal_load(&wbase[(size_t)(k0 + khalf + 0) * C_OUT + n]);
        b.y = __builtin_nontemporal_load(&wbase[(size_t)(k0 + khalf + 1) * C_OUT + n]);
        // D = A(16x4) * B(4x16) + C   (fp32)
        acc = __builtin_amdgcn_wmma_f32_16x16x4_f32(
                  /*neg_a=*/false, a, /*neg_b=*/false, b,
                  /*c_mod=*/(short)0, acc, /*reuse_a=*/false, /*reuse_b=*/false);
    }

    // ---- bias + store: vgpr i holds row M=i (lanes 0-15) / M=i+8 (lanes 16-31), N = lane%16
    const int   o     = wave * 16 + n;
    const float bv    = bias[(size_t)pos * C_OUT + o];
    const int   mbase = (lane >> 4) << 3;               // 0 or 8
    // 8 stores at a compile-time-constant stride from one base address
    float* obase = out + ((size_t)(mbase * C_OUT + o) * H_OUT + u) * W_OUT + v;
    constexpr size_t bstride = (size_t)C_OUT * H_OUT * W_OUT;   // 200704 elems, 802816 B
    #pragma unroll
    for (int i = 0; i < 8; ++i) {
        __builtin_nontemporal_store(acc[i] + bv, obase + (size_t)i * bstride);
    }
}

extern "C" void kernel_launch(void* const* d_in, const int* in_sizes, int n_in,
                              void* d_out, int out_size, void* d_ws, size_t ws_size,
                              hipStream_t stream) {
    const float* x    = (const float*)d_in[0];
    const float* wgt  = (const float*)d_in[1];
    const float* bias = (const float*)d_in[2];
    float*       out  = (float*)d_out;

    dim3 grid(H_OUT * W_OUT);   // 3136 positions
    dim3 block(128);            // 4 wave32s, one per 16-wide C_OUT tile
    hipLaunchKernelGGL(lcl_wmma_f32_kernel, grid, block, 0, stream, x, wgt, bias, out);
}